// RoIHeads_50062138802859
// MI455X (gfx1250) — compile-verified
//
#include <hip/hip_runtime.h>
#include <math.h>

#define N_ROWS    1000
#define N_CLASSES 81
#define N_FG      80
#define NPAD      1024
#define NDET      100

typedef __attribute__((ext_vector_type(2))) float v2f;
typedef __attribute__((ext_vector_type(8))) float v8f;

// ---------------------------------------------------------------------------
// Kernel 1: per-row softmax statistics (rowmax, 1/sum(exp)) over 81 classes.
// One wave32 per 16-row tile. Row sums of exp() are formed with
// V_WMMA_F32_16X16X4_F32 against an all-ones B matrix: D[m][n] = sum_k A[m][k],
// so every column of D holds the row sum (robust to K-permutation details).
// ---------------------------------------------------------------------------
__global__ __launch_bounds__(32)
void rh_softmax_stats_kernel(const float* __restrict__ logit,
                             float* __restrict__ rowmax,
                             float* __restrict__ invsum) {
    const int lane = threadIdx.x;
    const int tile = blockIdx.x;
    const int mrow = lane & 15;
    const int row  = tile * 16 + mrow;
    const bool ok  = row < N_ROWS;

    // Row max over 81 logits (lanes 0..15 compute, shared to 16..31).
    float mx = -3.402823466e+38f;
    if (lane < 16 && ok) {
        const float* p = logit + (size_t)row * N_CLASSES;
        for (int c = 0; c < N_CLASSES; ++c) mx = fmaxf(mx, p[c]);
    }
    mx = __shfl(mx, lane & 15, 32);
    if (!ok) mx = 0.0f;

    const float* rowp = logit + (size_t)(ok ? row : 0) * N_CLASSES;
    const int koff = (lane >= 16) ? 2 : 0;   // ISA A-layout: lanes>=16 hold K=2,3

    __shared__ float ldssum[16];

#if __has_builtin(__builtin_amdgcn_wmma_f32_16x16x4_f32)
    v8f acc = {0.f, 0.f, 0.f, 0.f, 0.f, 0.f, 0.f, 0.f};
    v2f bones; bones[0] = 1.0f; bones[1] = 1.0f;
    for (int kc = 0; kc < 20; ++kc) {
        const int k0 = kc * 4 + koff;
        v2f a;
        a[0] = ok ? expf(rowp[k0]     - mx) : 0.0f;
        a[1] = ok ? expf(rowp[k0 + 1] - mx) : 0.0f;
        acc = __builtin_amdgcn_wmma_f32_16x16x4_f32(
            /*neg_a=*/false, a, /*neg_b=*/false, bones,
            /*c_mod=*/(short)0, acc, /*reuse_a=*/false, /*reuse_b=*/false);
    }
    // D layout: VGPR r -> M=r at lanes 0-15 (N=lane), M=8+r at lanes 16-31.
    if (lane == 0)  { for (int r = 0; r < 8; ++r) ldssum[r]     = acc[r]; }
    if (lane == 16) { for (int r = 0; r < 8; ++r) ldssum[8 + r] = acc[r]; }
#else
    // Fallback: lane L and lane L+16 hold complementary K-halves of row L&15.
    float partial = 0.0f;
    for (int kc = 0; kc < 20; ++kc) {
        const int k0 = kc * 4 + koff;
        partial += ok ? expf(rowp[k0]     - mx) : 0.0f;
        partial += ok ? expf(rowp[k0 + 1] - mx) : 0.0f;
    }
    partial += __shfl_xor(partial, 16, 32);
    if (lane < 16) ldssum[lane] = partial;
#endif
    __syncthreads();

    if (lane < 16 && ok) {
        const float s = ldssum[lane] + expf(rowp[80] - mx);  // remainder k=80
        rowmax[row] = mx;
        invsum[row] = 1.0f / s;
    }
}

// ---------------------------------------------------------------------------
// Kernel 2: one workgroup per foreground class. Decode+filter, bitonic sort
// in LDS (stable descending score), single-wave greedy NMS with a register-
// resident 1024-bit alive mask, then write top-100 detections.
// ---------------------------------------------------------------------------
__global__ __launch_bounds__(512)
void rh_nms_class_kernel(const float* __restrict__ logit,
                         const float* __restrict__ boxreg,
                         const float* __restrict__ proposal,
                         const float* __restrict__ rowmax,
                         const float* __restrict__ invsum,
                         float* __restrict__ out) {
    const int k   = blockIdx.x;   // 0..79
    const int cls = k + 1;        // label / column in logits & deltas
    const int tid = threadIdx.x;

    __shared__ float bx[NPAD * 4];                 // unsorted decoded boxes
    __shared__ unsigned long long skey[NPAD];      // sort keys
    __shared__ float sx1[NPAD], sy1[NPAD], sx2[NPAD], sy2[NPAD];
    __shared__ float sar[NPAD], ssc[NPAD];         // sorted SoA + area + score
    __shared__ int   keeplist[NDET];
    __shared__ int   keepcnt;

    const float CLIPV = 4.135166556742356f;        // log(1000/16)

    // ---- Phase A: score + decode + validity, build sort keys --------------
    for (int n = tid; n < NPAD; n += 512) {
        float s, x1 = 0.f, y1 = 0.f, x2 = 0.f, y2 = 0.f;
        if (n < N_ROWS) {
            const float mx  = rowmax[n];
            const float inv = invsum[n];
            const float sc  = expf(logit[(size_t)n * N_CLASSES + cls] - mx) * inv;

            const float px1 = proposal[n * 4 + 0], py1 = proposal[n * 4 + 1];
            const float px2 = proposal[n * 4 + 2], py2 = proposal[n * 4 + 3];
            const float w  = px2 - px1, h = py2 - py1;
            const float cx = px1 + 0.5f * w, cy = py1 + 0.5f * h;

            const float* d = boxreg + (size_t)n * (N_CLASSES * 4) + 4 * cls;
            const float dx = d[0] / 10.0f, dy = d[1] / 10.0f;
            const float dw = fminf(d[2] / 5.0f, CLIPV);
            const float dh = fminf(d[3] / 5.0f, CLIPV);

            const float pcx = dx * w + cx, pcy = dy * h + cy;
            const float pw  = expf(dw) * w, ph = expf(dh) * h;
            x1 = fminf(fmaxf(pcx - 0.5f * pw, 0.f), 800.f);
            y1 = fminf(fmaxf(pcy - 0.5f * ph, 0.f), 800.f);
            x2 = fminf(fmaxf(pcx + 0.5f * pw, 0.f), 800.f);
            y2 = fminf(fmaxf(pcy + 0.5f * ph, 0.f), 800.f);

            const bool valid = (sc >= 0.05f) && ((x2 - x1) >= 1.0f) && ((y2 - y1) >= 1.0f);
            s = valid ? sc : -1.0f;
        } else {
            s = -2.0f;                              // padding sorts last
        }
        bx[n * 4 + 0] = x1; bx[n * 4 + 1] = y1;
        bx[n * 4 + 2] = x2; bx[n * 4 + 3] = y2;
        // Monotone float->uint, inverted for descending; index = stable tiebreak.
        const unsigned u = __float_as_uint(s);
        const unsigned m = (u & 0x80000000u) ? ~u : (u | 0x80000000u);
        skey[n] = ((unsigned long long)(~m) << 32) | (unsigned)n;
    }
    __syncthreads();

    // ---- Phase B: bitonic sort ascending (=> score descending, idx asc) ---
    for (unsigned kk = 2; kk <= NPAD; kk <<= 1) {
        for (unsigned j = kk >> 1; j > 0; j >>= 1) {
            for (unsigned i = tid; i < NPAD; i += 512) {
                const unsigned l = i ^ j;
                if (l > i) {
                    const unsigned long long a = skey[i], b = skey[l];
                    const bool up = ((i & kk) == 0);
                    if (up ? (a > b) : (a < b)) { skey[i] = b; skey[l] = a; }
                }
            }
            __syncthreads();
        }
    }

    // ---- Phase C: gather sorted SoA -------------------------------------
    for (int p = tid; p < NPAD; p += 512) {
        const unsigned long long key = skey[p];
        const unsigned n = (unsigned)(key & 0xFFFFFFFFull);
        const unsigned m = ~((unsigned)(key >> 32));
        const unsigned u = (m & 0x80000000u) ? (m ^ 0x80000000u) : ~m;
        const float x1 = bx[n * 4 + 0], y1 = bx[n * 4 + 1];
        const float x2 = bx[n * 4 + 2], y2 = bx[n * 4 + 3];
        sx1[p] = x1; sy1[p] = y1; sx2[p] = x2; sy2[p] = y2;
        sar[p] = (x2 - x1) * (y2 - y1);
        ssc[p] = __uint_as_float(u);
    }
    __syncthreads();

    // ---- Phase D: greedy NMS, single wave32, no barriers ------------------
    // Lane L owns alive bit w for box j = L + 32*w (conflict-free LDS strides).
    if (tid < 32) {
        const int lane = tid;
        unsigned aw = 0;
        for (int w = 0; w < 32; ++w) {
            const int j = lane + 32 * w;
            if (ssc[j] > -0.5f) aw |= (1u << w);    // alive iff valid
        }
        int cnt = 0;
        for (int i = 0; i < N_ROWS; ++i) {
            const int src = i & 31, bitp = i >> 5;
            const unsigned wi = (unsigned)__shfl((int)aw, src, 32); // uniform
            if (((wi >> bitp) & 1u) == 0) continue;  // suppressed or invalid
            if (lane == 0) keeplist[cnt] = i;
            ++cnt;
            if (cnt >= NDET) break;                  // top-100 = first 100 kept
            const float X1 = sx1[i], Y1 = sy1[i], X2 = sx2[i], Y2 = sy2[i];
            const float A = sar[i];
            unsigned sup = 0;
            for (int w = 0; w < 32; ++w) {
                const int j = lane + 32 * w;
                const float iw = fmaxf(fminf(X2, sx2[j]) - fmaxf(X1, sx1[j]), 0.f);
                const float ih = fmaxf(fminf(Y2, sy2[j]) - fmaxf(Y1, sy1[j]), 0.f);
                const float inter = iw * ih;
                const float iou = inter / (A + sar[j] - inter + 1e-9f);
                if ((j > i) && (iou > 0.5f)) sup |= (1u << w);
            }
            aw &= ~sup;
        }
        if (lane == 0) keepcnt = cnt;
    }
    __syncthreads();

    // ---- Phase E: emit outputs (boxes | scores | labels | valid) ----------
    if (tid < NDET) {
        const int  cnt = keepcnt;
        const bool v   = tid < cnt;
        const int  p   = v ? keeplist[tid] : 0;
        const size_t oi = (size_t)k * NDET + tid;
        float* ob = out;                         // [80,100,4]
        float* os = out + N_FG * NDET * 4;       // [80,100]
        float* ol = os + N_FG * NDET;            // [80,100]
        float* ov = ol + N_FG * NDET;            // [80,100]
        ob[oi * 4 + 0] = v ? sx1[p] : 0.f;
        ob[oi * 4 + 1] = v ? sy1[p] : 0.f;
        ob[oi * 4 + 2] = v ? sx2[p] : 0.f;
        ob[oi * 4 + 3] = v ? sy2[p] : 0.f;
        os[oi] = v ? ssc[p] : 0.f;
        ol[oi] = v ? (float)cls : 0.f;
        ov[oi] = v ? 1.f : 0.f;
    }
}

extern "C" void kernel_launch(void* const* d_in, const int* in_sizes, int n_in,
                              void* d_out, int out_size, void* d_ws, size_t ws_size,
                              hipStream_t stream) {
    const float* logit    = (const float*)d_in[0];
    const float* boxreg   = (const float*)d_in[1];
    const float* proposal = (const float*)d_in[2];
    float* out    = (float*)d_out;
    float* rowmax = (float*)d_ws;            // 1000 floats
    float* invsum = rowmax + N_ROWS;         // 1000 floats (8 KB total)

    rh_softmax_stats_kernel<<<(N_ROWS + 15) / 16, 32, 0, stream>>>(logit, rowmax, invsum);
    rh_nms_class_kernel<<<N_FG, 512, 0, stream>>>(logit, boxreg, proposal,
                                                  rowmax, invsum, out);
}